// PostProcess_80994493268398
// MI455X (gfx1250) — compile-verified
//
#include <hip/hip_runtime.h>
#include <hip/hip_bf16.h>

typedef unsigned int       uint32;
typedef unsigned long long u64;

#define NBINS 8192   // 13-bit MSB radix bins
#define CAP   4096   // candidate cap per image (expected ~1K)

typedef float v2f __attribute__((ext_vector_type(2)));
typedef float v8f __attribute__((ext_vector_type(8)));

// Monotonic float -> u32 key: order(key) == order(float) ascending.
__device__ __forceinline__ uint32 f2key(float f) {
    uint32 u = __float_as_uint(f);
    return (u & 0x80000000u) ? ~u : (u | 0x80000000u);
}

__global__ void k_zero(uint32* __restrict__ p, int n) {
    int i = blockIdx.x * blockDim.x + threadIdx.x;
    if (i < n) p[i] = 0u;
}

// Pass 1: per-image 8192-bin histogram of key MSBs (LDS accumulate, global flush).
__global__ void k_hist(const float* __restrict__ logits, uint32* __restrict__ hist,
                       int perImage, int n4) {
    int b = blockIdx.y;
    __shared__ uint32 lh[NBINS];
    for (int i = threadIdx.x; i < NBINS; i += blockDim.x) lh[i] = 0u;
    __syncthreads();

    const float4* src = (const float4*)(logits + (size_t)b * perImage);
    int step = gridDim.x * blockDim.x;
    for (int i = blockIdx.x * blockDim.x + threadIdx.x; i < n4; i += step) {
        if (i + 2 * step < n4) __builtin_prefetch(src + i + 2 * step, 0, 0);
        float4 v = src[i];
        atomicAdd(&lh[f2key(v.x) >> 19], 1u);
        atomicAdd(&lh[f2key(v.y) >> 19], 1u);
        atomicAdd(&lh[f2key(v.z) >> 19], 1u);
        atomicAdd(&lh[f2key(v.w) >> 19], 1u);
    }
    int rem = perImage - n4 * 4;
    if (blockIdx.x == 0 && (int)threadIdx.x < rem) {
        float f = logits[(size_t)b * perImage + n4 * 4 + threadIdx.x];
        atomicAdd(&lh[f2key(f) >> 19], 1u);
    }
    __syncthreads();
    for (int i = threadIdx.x; i < NBINS; i += blockDim.x)
        if (lh[i]) atomicAdd(&hist[(size_t)b * NBINS + i], lh[i]);
}

// Pass 2: find threshold bin T such that count of elements in bins >= T reaches k.
__global__ void k_thresh(const uint32* __restrict__ hist, uint32* __restrict__ thrBin, int k) {
    int b = blockIdx.x;
    if (threadIdx.x == 0) {
        const uint32* h = hist + (size_t)b * NBINS;
        uint32 cum = 0; uint32 t = 0;
        for (int i = NBINS - 1; i >= 0; --i) {
            cum += h[i];
            if (cum >= (uint32)k) { t = (uint32)i; break; }
        }
        thrBin[b] = t;
    }
}

// Pass 3: collect all candidates with bin >= T as (key<<32 | ~idx) records.
__global__ void k_collect(const float* __restrict__ logits, const uint32* __restrict__ thrBin,
                          uint32* __restrict__ candCnt, u64* __restrict__ cand,
                          int perImage, int n4) {
    int b = blockIdx.y;
    uint32 T = thrBin[b];
    const float4* src = (const float4*)(logits + (size_t)b * perImage);
    int step = gridDim.x * blockDim.x;
    for (int i = blockIdx.x * blockDim.x + threadIdx.x; i < n4; i += step) {
        if (i + 2 * step < n4) __builtin_prefetch(src + i + 2 * step, 0, 0);
        float4 v = src[i];
        float vals[4] = {v.x, v.y, v.z, v.w};
#pragma unroll
        for (int c = 0; c < 4; ++c) {
            uint32 key = f2key(vals[c]);
            if ((key >> 19) >= T) {
                uint32 pos = atomicAdd(&candCnt[b], 1u);
                if (pos < CAP)
                    cand[(size_t)b * CAP + pos] =
                        ((u64)key << 32) | (u64)(0xFFFFFFFFu - (uint32)(i * 4 + c));
            }
        }
    }
    int rem = perImage - n4 * 4;
    if (blockIdx.x == 0 && (int)threadIdx.x < rem) {
        int idx = n4 * 4 + threadIdx.x;
        uint32 key = f2key(logits[(size_t)b * perImage + idx]);
        if ((key >> 19) >= T) {
            uint32 pos = atomicAdd(&candCnt[b], 1u);
            if (pos < CAP)
                cand[(size_t)b * CAP + pos] = ((u64)key << 32) | (u64)(0xFFFFFFFFu - (uint32)idx);
        }
    }
}

// Pass 4: per image: bitonic sort candidates (desc), emit scores/labels, and
// compute scaled xyxy boxes with V_WMMA_F32_16X16X4_F32 (16 boxes per tile).
__global__ __launch_bounds__(1024) void k_topk(const float* __restrict__ boxes,
                                               const int* __restrict__ tsz,
                                               float* __restrict__ out,
                                               const u64* __restrict__ cand,
                                               const uint32* __restrict__ candCnt,
                                               int B, int Q, int C, int K) {
    int b = blockIdx.x;
    int tid = threadIdx.x;
    __shared__ u64 arr[CAP];
    __shared__ int qidx_s[512];

    uint32 cnt = candCnt[b];
    if (cnt > CAP) cnt = CAP;
    for (int t = tid; t < CAP; t += blockDim.x)
        arr[t] = (t < (int)cnt) ? cand[(size_t)b * CAP + t] : 0ull;
    __syncthreads();

    // Bitonic sort, descending (padding zeros sink to the bottom).
    for (int size = 2; size <= CAP; size <<= 1) {
        for (int stride = size >> 1; stride > 0; stride >>= 1) {
            for (int t = tid; t < CAP; t += blockDim.x) {
                int j = t ^ stride;
                if (j > t) {
                    u64 x = arr[t], y = arr[j];
                    bool desc = ((t & size) == 0);
                    if (desc ? (x < y) : (x > y)) { arr[t] = y; arr[j] = x; }
                }
            }
            __syncthreads();
        }
    }

    // Scores + labels + query indices for the top K.
    for (int r = tid; r < K; r += blockDim.x) {
        u64 v = arr[r];
        float score = 0.f; int lab = 0, q = 0;
        if (v != 0ull) {
            uint32 key = (uint32)(v >> 32);
            uint32 idx = 0xFFFFFFFFu - (uint32)(v & 0xFFFFFFFFull);
            uint32 u = (key & 0x80000000u) ? (key & 0x7FFFFFFFu) : ~key;
            float logit = __uint_as_float(u);
            score = (logit >= 0.f) ? (1.f / (1.f + __expf(-logit)))
                                   : (__expf(logit) / (1.f + __expf(logit)));
            q   = (int)(idx / (uint32)C);
            lab = (int)(idx % (uint32)C);
        }
        out[(size_t)b * K + r] = score;
        out[(size_t)B * K + (size_t)b * K + r] = (float)lab;
        qidx_s[r] = q;
    }
    __syncthreads();

    // Box transform via WMMA: D(16x16) = A(16 boxes x 4 comps) * Bmat(4x16).
    // Bmat cols 0..3 fuse cxcywh->xyxy with per-image (W,H) scaling.
    float Wd = (float)tsz[b * 2 + 1];  // target_sizes = (h, w)
    float Hd = (float)tsz[b * 2 + 0];
    int wave = tid >> 5, lane = tid & 31;
    int nwaves = blockDim.x >> 5;
    int tiles = (K + 15) >> 4;
    for (int tile = wave; tile < tiles; tile += nwaves) {
        int m = lane & 15;
        int g = tile * 16 + m;
        // A layout (f32 16x4): lanes 0-15 hold K=0 (v0) / K=1 (v1) for row M=lane;
        //                      lanes 16-31 hold K=2 / K=3 for row M=lane-16.
        v2f a; a.x = 0.f; a.y = 0.f;
        if (g < K) {
            const float* bp = boxes + ((size_t)b * Q + qidx_s[g]) * 4;
            if (lane < 16) { a.x = bp[0]; a.y = bp[1]; }                    // xc, yc
            else           { a.x = fmaxf(bp[2], 0.f); a.y = fmaxf(bp[3], 0.f); } // w+, h+
        }
        // B layout (f32 4x16): lanes 0-15 hold rows K=0 (v0) / K=1 (v1), N=lane;
        //                      lanes 16-31 hold rows K=2 / K=3, N=lane-16.
        v2f bm; bm.x = 0.f; bm.y = 0.f;
        int n = lane & 15;
        if (lane < 16) {
            if (n == 0 || n == 2) bm.x = Wd;          // row0: xc coeff
            if (n == 1 || n == 3) bm.y = Hd;          // row1: yc coeff
        } else {
            if (n == 0) bm.x = -0.5f * Wd; else if (n == 2) bm.x = 0.5f * Wd; // row2: w
            if (n == 1) bm.y = -0.5f * Hd; else if (n == 3) bm.y = 0.5f * Hd; // row3: h
        }
        v8f c = {0.f, 0.f, 0.f, 0.f, 0.f, 0.f, 0.f, 0.f};
        c = __builtin_amdgcn_wmma_f32_16x16x4_f32(false, a, false, bm,
                                                  (short)0, c, false, false);
        // D layout: vgpr j -> lanes 0-15: (M=j, N=lane); lanes 16-31: (M=j+8, N=lane-16)
        if (n < 4) {
            int half = lane >> 4;
#pragma unroll
            for (int j = 0; j < 8; ++j) {
                int g2 = tile * 16 + half * 8 + j;
                if (g2 < K)
                    out[2 * (size_t)B * K + ((size_t)b * K + g2) * 4 + n] = c[j];
            }
        }
    }
}

extern "C" void kernel_launch(void* const* d_in, const int* in_sizes, int n_in,
                              void* d_out, int out_size, void* d_ws, size_t ws_size,
                              hipStream_t stream) {
    const float* logits = (const float*)d_in[0];
    const float* boxes  = (const float*)d_in[1];
    const int*   tsz    = (const int*)d_in[2];

    int B = in_sizes[2] / 2;                 // 64
    int Q = in_sizes[1] / (B * 4);           // 900
    int C = in_sizes[0] / (B * Q);           // 1203
    int K = out_size / (B * 6);              // 300
    int perImage = Q * C;
    int n4 = perImage / 4;

    uint32* hist    = (uint32*)d_ws;                     // B*NBINS
    uint32* thrBin  = hist + (size_t)B * NBINS;          // B
    uint32* candCnt = thrBin + B;                        // B
    size_t candOff  = (((size_t)B * NBINS + 2 * (size_t)B) * 4 + 7) & ~(size_t)7;
    u64*    cand    = (u64*)((char*)d_ws + candOff);     // B*CAP u64

    int zn = B * NBINS + 2 * B;
    k_zero<<<(zn + 255) / 256, 256, 0, stream>>>(hist, zn);

    dim3 gstream(160, B);
    k_hist<<<gstream, 256, 0, stream>>>(logits, hist, perImage, n4);
    k_thresh<<<B, 32, 0, stream>>>(hist, thrBin, K);
    k_collect<<<gstream, 256, 0, stream>>>(logits, thrBin, candCnt, cand, perImage, n4);
    k_topk<<<B, 1024, 0, stream>>>(boxes, tsz, (float*)d_out, cand, candCnt, B, Q, C, K);
}